// WeightedGCN_46626164965918
// MI455X (gfx1250) — compile-verified
//
#include <hip/hip_runtime.h>
#include <hip/hip_bf16.h>

#define NN 100000   // nodes
#define NE 600000   // edges
#define NG 64       // graphs
#define DD 128      // feature dim
#define LN_EPS 1e-5f

typedef __attribute__((ext_vector_type(2))) float v2f;
typedef __attribute__((ext_vector_type(8))) float v8f;

// device-scope relaxed fp32 atomic add -> should lower to global_atomic_add_f32
__device__ __forceinline__ void atomAddF(float* p, float v) {
    __hip_atomic_fetch_add(p, v, __ATOMIC_RELAXED, __HIP_MEMORY_SCOPE_AGENT);
}

// ---------------------------------------------------------------- utilities
__global__ void wgcn_zero_kernel(float* __restrict__ p, long n) {
    long i = (long)blockIdx.x * blockDim.x + threadIdx.x;
    if (i < n) p[i] = 0.0f;
}

// deg[col[e]] += ew[e]
__global__ void wgcn_deg_kernel(const int* __restrict__ ei, const float* __restrict__ ew,
                                float* __restrict__ deg) {
    long e = (long)blockIdx.x * blockDim.x + threadIdx.x;
    if (e < NE) {
        int c = ei[NE + e];
        atomAddF(&deg[c], ew[e]);
    }
}

// dis[n] = rsqrt(deg[n] + 1)   (self-loop weight 1)
__global__ void wgcn_dis_kernel(const float* __restrict__ deg, float* __restrict__ dis) {
    long n = (long)blockIdx.x * blockDim.x + threadIdx.x;
    if (n < NN) dis[n] = rsqrtf(deg[n] + 1.0f);
}

// ------------------------------------------------- embedding gather + LayerNorm
// one wave32 per node, 4 floats per lane
__global__ __launch_bounds__(256)
void wgcn_embed_ln_kernel(const int* __restrict__ xidx, const float* __restrict__ emb,
                          const float* __restrict__ s, const float* __restrict__ b,
                          float* __restrict__ out) {
    long node = ((long)blockIdx.x * blockDim.x + threadIdx.x) >> 5;
    int lane = threadIdx.x & 31;
    if (node >= NN) return;
    int idx = xidx[node];
    float4 v;
    if (idx == 0) { v.x = v.y = v.z = v.w = 0.0f; }  // padding_idx=0 -> zero row
    else v = *(const float4*)(emb + (long)idx * DD + lane * 4);

    float sum = v.x + v.y + v.z + v.w;
    #pragma unroll
    for (int m = 16; m >= 1; m >>= 1) sum += __shfl_xor(sum, m, 32);
    float mean = sum * (1.0f / DD);
    float dx0 = v.x - mean, dx1 = v.y - mean, dx2 = v.z - mean, dx3 = v.w - mean;
    float ss = dx0*dx0 + dx1*dx1 + dx2*dx2 + dx3*dx3;
    #pragma unroll
    for (int m = 16; m >= 1; m >>= 1) ss += __shfl_xor(ss, m, 32);
    float inv = rsqrtf(ss * (1.0f / DD) + LN_EPS);
    float4 sc = *(const float4*)(s + lane * 4);
    float4 bb = *(const float4*)(b + lane * 4);
    float4 o;
    o.x = dx0 * inv * sc.x + bb.x;
    o.y = dx1 * inv * sc.y + bb.y;
    o.z = dx2 * inv * sc.z + bb.z;
    o.w = dx3 * inv * sc.w + bb.w;
    *(float4*)(out + node * DD + lane * 4) = o;
}

// -------------------------------------------------------------- WMMA GEMM
// H[NN,128] = A[NN,128] @ W[128,128], fp32 via V_WMMA_F32_16X16X4_F32.
// Fused epilogue also writes AGG = H * dis^2 (GCN self-loop term), saving a
// full 51 MB re-read pass per layer.
// Block = 256 threads (8 waves) per 16-row tile; wave w owns cols [16w,16w+16).
// W staged in LDS in two 64-row halves; +4 float padding avoids bank conflicts.
__global__ __launch_bounds__(256)
void wgcn_gemm_wmma_kernel(const float* __restrict__ A, const float* __restrict__ W,
                           const float* __restrict__ dis,
                           float* __restrict__ H, float* __restrict__ AGG) {
    __shared__ float sW[64 * 132];
    __shared__ float sA[16 * 132];
    const int tid = threadIdx.x;
    const long m0 = (long)blockIdx.x * 16;

    // stage 16x128 A tile
    for (int i = tid; i < 16 * DD; i += 256) {
        int m = i >> 7, k = i & 127;
        sA[m * 132 + k] = A[(m0 + m) * DD + k];
    }

    const int wave = tid >> 5, lane = tid & 31;
    const int wcol = wave << 4;            // this wave's 16-col strip
    const int arow = lane & 15;            // A: M = lane&15
    const int koff = (lane >> 4) << 1;     // A: K pair 0-1 (lanes 0-15) / 2-3 (16-31)
    const int bn   = wcol + (lane & 15);   // B: N = lane&15 within strip

    v8f acc = {0.0f, 0.0f, 0.0f, 0.0f, 0.0f, 0.0f, 0.0f, 0.0f};

    for (int half = 0; half < 2; ++half) {
        const int k0 = half << 6;          // 0 or 64
        __syncthreads();                   // protect sW reuse across halves
        for (int i = tid; i < 64 * DD; i += 256) {
            int k = i >> 7, n = i & 127;
            sW[k * 132 + n] = W[(k0 + k) * DD + n];
        }
        __syncthreads();
        #pragma unroll
        for (int kb = 0; kb < 64; kb += 4) {
            v2f a, b;
            a[0] = sA[arow * 132 + k0 + kb + koff];
            a[1] = sA[arow * 132 + k0 + kb + koff + 1];
            b[0] = sW[(kb + koff) * 132 + bn];
            b[1] = sW[(kb + koff + 1) * 132 + bn];
            acc = __builtin_amdgcn_wmma_f32_16x16x4_f32(
                false, a, false, b, (short)0, acc, false, false);
        }
    }

    // C/D layout: VGPR r -> M=r (lanes 0-15) / M=r+8 (lanes 16-31), N=lane&15
    const int crow = (lane >> 4) << 3;
    const int cn = lane & 15;
    #pragma unroll
    for (int r = 0; r < 8; ++r) {
        long row = m0 + crow + r;
        float hv = acc[r];
        float d  = dis[row];
        H  [row * DD + wcol + cn] = hv;
        AGG[row * DD + wcol + cn] = hv * d * d;   // self-loop init
    }
}

// one wave per edge: agg[col] += h[row] * (dis[row]*ew*dis[col])
__global__ __launch_bounds__(256)
void wgcn_scatter_kernel(const int* __restrict__ ei, const float* __restrict__ ew,
                         const float* __restrict__ dis, const float* __restrict__ h,
                         float* __restrict__ agg) {
    long e = ((long)blockIdx.x * blockDim.x + threadIdx.x) >> 5;
    int lane = threadIdx.x & 31;
    if (e >= NE) return;
    int r = ei[e];
    int c = ei[NE + e];
    float nw = dis[r] * ew[e] * dis[c];
    float4 hv = *(const float4*)(h + (long)r * DD + lane * 4);
    float* dst = agg + (long)c * DD + lane * 4;
    atomAddF(dst + 0, hv.x * nw);
    atomAddF(dst + 1, hv.y * nw);
    atomAddF(dst + 2, hv.z * nw);
    atomAddF(dst + 3, hv.w * nw);
}

// y = relu(agg + bias); out = LayerNorm(y) with (s,b). One wave per node.
__global__ __launch_bounds__(256)
void wgcn_bias_relu_ln_kernel(const float* __restrict__ agg, const float* __restrict__ bias,
                              const float* __restrict__ s, const float* __restrict__ b,
                              float* __restrict__ out) {
    long node = ((long)blockIdx.x * blockDim.x + threadIdx.x) >> 5;
    int lane = threadIdx.x & 31;
    if (node >= NN) return;
    float4 v = *(const float4*)(agg + node * DD + lane * 4);
    float4 bi = *(const float4*)(bias + lane * 4);
    v.x = fmaxf(v.x + bi.x, 0.0f);
    v.y = fmaxf(v.y + bi.y, 0.0f);
    v.z = fmaxf(v.z + bi.z, 0.0f);
    v.w = fmaxf(v.w + bi.w, 0.0f);
    float sum = v.x + v.y + v.z + v.w;
    #pragma unroll
    for (int m = 16; m >= 1; m >>= 1) sum += __shfl_xor(sum, m, 32);
    float mean = sum * (1.0f / DD);
    float dx0 = v.x - mean, dx1 = v.y - mean, dx2 = v.z - mean, dx3 = v.w - mean;
    float ss = dx0*dx0 + dx1*dx1 + dx2*dx2 + dx3*dx3;
    #pragma unroll
    for (int m = 16; m >= 1; m >>= 1) ss += __shfl_xor(ss, m, 32);
    float inv = rsqrtf(ss * (1.0f / DD) + LN_EPS);
    float4 sc = *(const float4*)(s + lane * 4);
    float4 bb = *(const float4*)(b + lane * 4);
    float4 o;
    o.x = dx0 * inv * sc.x + bb.x;
    o.y = dx1 * inv * sc.y + bb.y;
    o.z = dx2 * inv * sc.z + bb.z;
    o.w = dx3 * inv * sc.w + bb.w;
    *(float4*)(out + node * DD + lane * 4) = o;
}

// y = relu(agg + bias); pool[batch[n]] += y; cnt[batch[n]] += 1. One wave per node.
__global__ __launch_bounds__(256)
void wgcn_pool_kernel(const float* __restrict__ agg, const float* __restrict__ bias,
                      const int* __restrict__ batch, float* __restrict__ pool,
                      float* __restrict__ cnt) {
    long node = ((long)blockIdx.x * blockDim.x + threadIdx.x) >> 5;
    int lane = threadIdx.x & 31;
    if (node >= NN) return;
    int g = batch[node];
    float4 v = *(const float4*)(agg + node * DD + lane * 4);
    float4 bi = *(const float4*)(bias + lane * 4);
    float* dst = pool + (long)g * DD + lane * 4;
    atomAddF(dst + 0, fmaxf(v.x + bi.x, 0.0f));
    atomAddF(dst + 1, fmaxf(v.y + bi.y, 0.0f));
    atomAddF(dst + 2, fmaxf(v.z + bi.z, 0.0f));
    atomAddF(dst + 3, fmaxf(v.w + bi.w, 0.0f));
    if (lane == 0) atomAddF(&cnt[g], 1.0f);
}

// tiny MLP head: one block, 64 threads, one graph per thread
__global__ __launch_bounds__(64)
void wgcn_head_kernel(const float* __restrict__ pool, const float* __restrict__ cnt,
                      const float* __restrict__ w1, const float* __restrict__ b1,
                      const float* __restrict__ w2, const float* __restrict__ b2,
                      float* __restrict__ out) {
    __shared__ float sg[NG][DD + 1];
    int g = threadIdx.x;
    float invc = 1.0f / fmaxf(cnt[g], 1.0f);
    for (int d = 0; d < DD; ++d) sg[g][d] = pool[(long)g * DD + d] * invc;
    __syncthreads();
    float o0 = b2[0], o1 = b2[1];
    for (int j = 0; j < 64; ++j) {
        float hj = b1[j];
        for (int d = 0; d < DD; ++d) hj += sg[g][d] * w1[d * 64 + j];
        hj = fmaxf(hj, 0.0f);
        o0 += hj * w2[j * 2 + 0];
        o1 += hj * w2[j * 2 + 1];
    }
    out[g * 2 + 0] = o0;
    out[g * 2 + 1] = o1;
}

extern "C" void kernel_launch(void* const* d_in, const int* in_sizes, int n_in,
                              void* d_out, int out_size, void* d_ws, size_t ws_size,
                              hipStream_t stream) {
    const int*   x_idx = (const int*)  d_in[0];
    const int*   ei    = (const int*)  d_in[1];   // [2, E]
    const float* ew    = (const float*)d_in[2];
    const int*   batch = (const int*)  d_in[3];
    const float* emb   = (const float*)d_in[4];
    const float* ln1s  = (const float*)d_in[5];
    const float* ln1b  = (const float*)d_in[6];
    const float* w1    = (const float*)d_in[7];
    const float* b1    = (const float*)d_in[8];
    const float* ln2s  = (const float*)d_in[9];
    const float* ln2b  = (const float*)d_in[10];
    const float* w2    = (const float*)d_in[11];
    const float* b2    = (const float*)d_in[12];
    const float* mw1   = (const float*)d_in[13];
    const float* mb1   = (const float*)d_in[14];
    const float* mw2   = (const float*)d_in[15];
    const float* mb2   = (const float*)d_in[16];
    float* out = (float*)d_out;

    // workspace layout (floats): three [N,128] arrays + deg/dis/pool/cnt
    float* ws   = (float*)d_ws;
    float* xln  = ws;                               // N*128
    float* h    = xln + (size_t)NN * DD;            // N*128
    float* agg  = h   + (size_t)NN * DD;            // N*128
    float* deg  = agg + (size_t)NN * DD;            // N
    float* dis  = deg + NN;                         // N
    float* pool = dis + NN;                         // G*128
    float* cnt  = pool + (size_t)NG * DD;           // G

    const int B = 256;
    const long zcount = (long)NN + NN + (long)NG * DD + NG;  // deg..cnt contiguous
    wgcn_zero_kernel<<<(int)((zcount + B - 1) / B), B, 0, stream>>>(deg, zcount);

    wgcn_deg_kernel<<<(NE + B - 1) / B, B, 0, stream>>>(ei, ew, deg);
    wgcn_dis_kernel<<<(NN + B - 1) / B, B, 0, stream>>>(deg, dis);

    const int waveGrid = (NN + 7) / 8;     // 8 waves per 256-thread block
    wgcn_embed_ln_kernel<<<waveGrid, B, 0, stream>>>(x_idx, emb, ln1s, ln1b, xln);

    const int gemmGrid = NN / 16;          // N % 16 == 0
    const int edgeGrid = (NE + 7) / 8;

    // ---- layer 1 (GEMM epilogue writes both h and agg = h*dis^2)
    wgcn_gemm_wmma_kernel<<<gemmGrid, B, 0, stream>>>(xln, w1, dis, h, agg);
    wgcn_scatter_kernel<<<edgeGrid, B, 0, stream>>>(ei, ew, dis, h, agg);
    wgcn_bias_relu_ln_kernel<<<waveGrid, B, 0, stream>>>(agg, b1, ln2s, ln2b, xln);

    // ---- layer 2
    wgcn_gemm_wmma_kernel<<<gemmGrid, B, 0, stream>>>(xln, w2, dis, h, agg);
    wgcn_scatter_kernel<<<edgeGrid, B, 0, stream>>>(ei, ew, dis, h, agg);

    // ---- pool + head
    wgcn_pool_kernel<<<waveGrid, B, 0, stream>>>(agg, b2, batch, pool, cnt);
    wgcn_head_kernel<<<1, NG, 0, stream>>>(pool, cnt, mw1, mb1, mw2, mb2, out);
}